// AMXReversibleLayer_87385404605275
// MI455X (gfx1250) — compile-verified
//
#include <hip/hip_runtime.h>
#include <stdint.h>

typedef __attribute__((ext_vector_type(2))) float v2f;
typedef __attribute__((ext_vector_type(8))) float v8f;

#define HD 128            // hidden dim
#define ROWLEN 256        // floats per x/out row (2*HD)
#define W_PAIR_STRIDE 288 // dwords per packed W pair-row in LDS (256 + 32 pad)
#define A_STRIDE 132      // dwords per x1 row in LDS (128 + 4 pad)
#define ROWS_PER_WAVE 16
#define WAVES_PER_BLOCK 8
#define ROWS_PER_BLOCK (ROWS_PER_WAVE * WAVES_PER_BLOCK)

// LDS budget: W packed = 64*288*4 = 73,728 B ; A tiles = 8*16*132*4 = 67,584 B
// total = 141,312 B (< 320 KB per WGP; two blocks can co-reside)
#define LDS_W_FLOATS (64 * W_PAIR_STRIDE)
#define LDS_A_FLOATS (WAVES_PER_BLOCK * ROWS_PER_WAVE * A_STRIDE)

static __device__ __forceinline__ void wait_asynccnt0() {
#if __has_builtin(__builtin_amdgcn_s_wait_asynccnt)
    __builtin_amdgcn_s_wait_asynccnt(0);
#else
    asm volatile("s_wait_asynccnt 0x0" ::: "memory");
#endif
}

extern "C" __global__ __launch_bounds__(256, 1)
void revnet_coupling_wmma_f32(const float* __restrict__ x,
                              const float* __restrict__ w,
                              float* __restrict__ out) {
    extern __shared__ float lds[];
    float* wLds = lds;                   // packed pair-row W
    float* aLds = lds + LDS_W_FLOATS;    // per-wave x1 tiles

    const int tid  = threadIdx.x;
    const int lane = tid & 31;
    const int wave = tid >> 5;
    const long long blockRow0 = (long long)blockIdx.x * ROWS_PER_BLOCK;
    const long long rowBase   = blockRow0 + (long long)wave * ROWS_PER_WAVE;

    // Per-lane LDS byte offset of this wave's x1 tile: generic LDS pointers carry
    // the LDS byte offset in their low 32 bits (flat aperture mapping, ISA 10.2).
    float* awp = aLds + wave * (ROWS_PER_WAVE * A_STRIDE);
    const unsigned ldsTileBase =
        (unsigned)(unsigned long long)(uintptr_t)awp + (unsigned)(lane * 16);

    // ---------- Phase 0: async-stage this wave's 16x128 x1 tile (global -> LDS) ----------
    // 32 lanes x b128 = 512 B per instruction = one full x1 half-row; ASYNCcnt-tracked,
    // streams under the W repack below with zero VGPR round-trip.
    #pragma unroll
    for (int r = 0; r < 16; ++r) {
        unsigned ldsA = ldsTileBase + (unsigned)(r * A_STRIDE * 4);
        const float* gp = x + (rowBase + r) * ROWLEN + lane * 4;
        asm volatile("global_load_async_to_lds_b128 %0, %1, off"
                     :
                     : "v"(ldsA), "v"((unsigned long long)(uintptr_t)gp)
                     : "memory");
    }

    // ---------- Phase 1: stage W (128x128 f32) into LDS, pair-row packed ----------
    // element (r, c) -> wLds[(r>>1)*288 + c*2 + (r&1)]  so each B operand is one b64.
    #pragma unroll
    for (int j = 0; j < 16; ++j) {
        int i = tid + j * 256;            // float4 index, 4096 total
        int f = i << 2;                   // flat float index
        int r = f >> 7;                   // row 0..127
        int c = f & 127;                  // col, multiple of 4
        float4 v = ((const float4*)w)[i]; // coalesced b128
        float* dst = wLds + (r >> 1) * W_PAIR_STRIDE + (r & 1);
        dst[(c + 0) * 2] = v.x;
        dst[(c + 1) * 2] = v.y;
        dst[(c + 2) * 2] = v.z;
        dst[(c + 3) * 2] = v.w;
    }

    // Drain async loads: x1 tile is now valid in LDS for this wave.
    wait_asynccnt0();

    // ---------- Phase 2: y1 = x1 passthrough, async LDS -> global (no VGPRs) ----------
    // Stores only READ LDS, so they overlap the GEMM below; S_ENDPGM's implicit
    // wait-idle guarantees completion before the kernel retires.
    #pragma unroll
    for (int r = 0; r < 16; ++r) {
        unsigned ldsA = ldsTileBase + (unsigned)(r * A_STRIDE * 4);
        float* gp = out + (rowBase + r) * ROWLEN + lane * 4;
        asm volatile("global_store_async_from_lds_b128 %0, %1, off"
                     :
                     : "v"((unsigned long long)(uintptr_t)gp), "v"(ldsA)
                     : "memory");
    }

    // ---------- Phase 3: seed accumulators with x2 so WMMA directly yields y2 ----------
    const int half = lane >> 4;   // K-pair selector within A/B operands
    const int lm   = lane & 15;   // M for A, N for B/C

    v8f acc[8];
    #pragma unroll
    for (int n = 0; n < 8; ++n) {
        #pragma unroll
        for (int v = 0; v < 8; ++v) {
            long long r = rowBase + v + half * 8;
            acc[n][v] = x[r * ROWLEN + HD + n * 16 + lm];
        }
    }

    __syncthreads();   // W visible to all waves

    // ---------- Phase 4: h = x1 @ W via V_WMMA_F32_16X16X4_F32 ----------
    const float* aBase = awp + lm * A_STRIDE + half * 2;

    #pragma unroll 2
    for (int k = 0; k < 32; ++k) {        // K = 128 in steps of 4
        v2f a = *(const v2f*)(aBase + 4 * k);
        const float* bBase = wLds + (2 * k + half) * W_PAIR_STRIDE + lm * 2;
        // Preload all 8 B operands into distinct registers so DS loads pipeline
        // under the WMMA chain (partial dscnt waits instead of full drains).
        v2f b[8];
        #pragma unroll
        for (int n = 0; n < 8; ++n)
            b[n] = *(const v2f*)(bBase + n * 32);
        #pragma unroll
        for (int n = 0; n < 8; ++n)
            acc[n] = __builtin_amdgcn_wmma_f32_16x16x4_f32(
                /*neg_a=*/false, a, /*neg_b=*/false, b[n],
                /*c_mod=*/(short)0, acc[n], /*reuse_a=*/false, /*reuse_b=*/false);
    }

    // ---------- Epilogue: store y2 (two contiguous 64B segments per b32 store) ----------
    #pragma unroll
    for (int n = 0; n < 8; ++n) {
        #pragma unroll
        for (int v = 0; v < 8; ++v) {
            long long r = rowBase + v + half * 8;
            out[r * ROWLEN + HD + n * 16 + lm] = acc[n][v];
        }
    }
}

extern "C" void kernel_launch(void* const* d_in, const int* in_sizes, int n_in,
                              void* d_out, int out_size, void* d_ws, size_t ws_size,
                              hipStream_t stream) {
    const float* x   = (const float*)d_in[0];  // [8, 32768, 256] f32
    const float* wgt = (const float*)d_in[1];  // [128, 128] f32
    float* out = (float*)d_out;                // [8, 32768, 256] f32

    const long long totalRows = 8LL * 32768LL;         // 262,144
    dim3 grid((unsigned)(totalRows / ROWS_PER_BLOCK)); // 2048 blocks
    dim3 block(256);                                   // 8 waves (wave32)
    size_t ldsBytes = (size_t)(LDS_W_FLOATS + LDS_A_FLOATS) * sizeof(float);

    hipLaunchKernelGGL(revnet_coupling_wmma_f32, grid, block, ldsBytes, stream,
                       x, wgt, out);
}